// SignatureTransform_82222853915369
// MI455X (gfx1250) — compile-verified
//
#include <hip/hip_runtime.h>

// Depth-4 path signature (Chen scan), d=8, path length 256, batch 128.
// One block (4 waves, 128 threads) per path. Level-4 accumulator c4 (64x64 f32)
// lives in V_WMMA_F32_16X16X4_F32 accumulator VGPRs across the whole scan;
// each step is one 64x20x64 f32 GEMM (K=17 padded to 20). Levels 1..3 in LDS.
// B is staged K-contiguous (transposed) so each WMMA fragment is one ds_load_b64.

typedef __attribute__((ext_vector_type(2))) float v2f;
typedef __attribute__((ext_vector_type(8))) float v8f;

#define DIMS   8
#define PLEN   256
#define NSTEP  255          // increments per path
#define KPAD   20           // K = 17 (8 + 1 + 8) padded to multiple of 4
#define OUT_PER 4680        // 8 + 64 + 512 + 4096

__global__ __launch_bounds__(128)
void sig_depth4_kernel(const float* __restrict__ path, float* __restrict__ out) {
    __shared__ float s_path[PLEN * DIMS];   // 8 KB: whole path, loaded once
    __shared__ float s_c1[DIMS];
    __shared__ float s_c2[DIMS * DIMS];
    __shared__ float s_c3[DIMS * DIMS * DIMS];
    __shared__ float s_e2[DIMS * DIMS];
    __shared__ float s_e3[DIMS * DIMS * DIMS];
    __shared__ float s_A [64 * KPAD];       // A[row][k]   : 64 x 20, K contiguous
    __shared__ float s_Bt[64 * KPAD];       // B_T[col][k] : 64 x 20, K contiguous

    const int tid  = threadIdx.x;           // 0..127
    const int lane = tid & 31;              // wave32
    const int wv   = tid >> 5;              // wave id 0..3 -> c4 row strip 16*wv
    const int p    = blockIdx.x;

    // ---- load path, zero carries ----
    for (int i = tid; i < PLEN * DIMS; i += 128)
        s_path[i] = path[p * (PLEN * DIMS) + i];
    if (tid < DIMS) s_c1[tid] = 0.0f;
    if (tid < DIMS * DIMS) s_c2[tid] = 0.0f;
    for (int i = tid; i < DIMS * DIMS * DIMS; i += 128) s_c3[i] = 0.0f;
    __syncthreads();

    // c4 accumulators: wave wv owns rows [16*wv, 16*wv+16), 4 column tiles.
    v8f acc0 = {}, acc1 = {}, acc2 = {}, acc3 = {};

    const int mrow = 16 * wv + (lane & 15);     // A row this lane feeds
    const int ncol = lane & 15;                 // column within a 16-wide tile
    const int kgrp = 2 * (lane >> 4);           // lanes>=16 take K upper half

    for (int t = 0; t < NSTEP; ++t) {
        // increment dx (broadcast LDS reads -> registers)
        float dx[DIMS];
#pragma unroll
        for (int i = 0; i < DIMS; ++i)
            dx[i] = s_path[(t + 1) * DIMS + i] - s_path[t * DIMS + i];

        // e2 = dx (x) dx / 2 ; e3 = dx^(x)3 / 6
        if (tid < 64) s_e2[tid] = dx[tid >> 3] * dx[tid & 7] * 0.5f;
        for (int i = tid; i < 512; i += 128) {
            int b = i >> 6, c = (i >> 3) & 7, d = i & 7;
            s_e3[i] = dx[b] * dx[c] * dx[d] * (1.0f / 6.0f);
        }
        __syncthreads();  // e2/e3 ready; c1..c3 still hold OLD values

        // ---- stage A (64 rows x 20 K), uses old c1,c2,c3 ----
        for (int i = tid; i < 64 * KPAD; i += 128) {
            int r = i / KPAD, k = i - r * KPAD;
            int a = r >> 3, b = r & 7;
            float v;
            if (k < 8)        v = s_c3[r * 8 + k];                       // c3 (x) e1
            else if (k == 8)  v = s_c2[r] + s_e2[r] * (1.0f / 6.0f);     // c2 (x) e2 + e4
            else if (k < 17)  v = (b == (k - 9)) ? s_c1[a] : 0.0f;       // c1 (x) e3
            else              v = 0.0f;                                  // K pad
            s_A[i] = v;
        }
        // ---- stage B transposed (64 cols x 20 K) ----
        for (int i = tid; i < 64 * KPAD; i += 128) {
            int col = i / KPAD, k = i - col * KPAD;
            float v;
            if (k < 8)        v = ((col >> 3) == k) ? dx[col & 7] : 0.0f; // E1
            else if (k == 8)  v = s_e2[col];
            else if (k < 17)  v = s_e3[(k - 9) * 64 + col];
            else              v = 0.0f;
            s_Bt[i] = v;
        }
        __syncthreads();  // A/B staged

        // ---- c4 += A @ B : 5 K-chunks of 4, 4 output tiles per wave ----
#pragma unroll
        for (int kk = 0; kk < KPAD / 4; ++kk) {
            const int kb = 4 * kk + kgrp;
            v2f af = *(const v2f*)&s_A [mrow * KPAD + kb];
            v2f b0 = *(const v2f*)&s_Bt[( 0 + ncol) * KPAD + kb];
            v2f b1 = *(const v2f*)&s_Bt[(16 + ncol) * KPAD + kb];
            v2f b2 = *(const v2f*)&s_Bt[(32 + ncol) * KPAD + kb];
            v2f b3 = *(const v2f*)&s_Bt[(48 + ncol) * KPAD + kb];
            acc0 = __builtin_amdgcn_wmma_f32_16x16x4_f32(false, af, false, b0, (short)0, acc0, false, false);
            acc1 = __builtin_amdgcn_wmma_f32_16x16x4_f32(false, af, false, b1, (short)0, acc1, false, false);
            acc2 = __builtin_amdgcn_wmma_f32_16x16x4_f32(false, af, false, b2, (short)0, acc2, false, false);
            acc3 = __builtin_amdgcn_wmma_f32_16x16x4_f32(false, af, false, b3, (short)0, acc3, false, false);
        }

        // ---- lower-level updates: compute c1/c2 new values in registers
        //      (old c1,c2 still needed by the in-place c3 update) ----
        float c2new = 0.0f, c1new = 0.0f;
        if (tid < 64) c2new = s_c2[tid] + s_e2[tid] + s_c1[tid >> 3] * dx[tid & 7];
        if (tid < DIMS) c1new = s_c1[tid] + dx[tid];
        // c3 += e3 + c1 (x) e2 + c2 (x) e1   (reads old c1,c2; writes only c3)
        for (int i = tid; i < 512; i += 128)
            s_c3[i] += s_e3[i] + s_c1[i >> 6] * s_e2[i & 63] + s_c2[i >> 3] * dx[i & 7];
        __syncthreads();  // all reads of old c1/c2 complete
        if (tid < 64) s_c2[tid] = c2new;
        if (tid < DIMS) s_c1[tid] = c1new;
        __syncthreads();  // commit; also fences e2/e3/A/B vs next-iter writes
    }

    // ---- write output: [c1 | c2 | c3 | c4], row-major flatten ----
    float* o = out + (size_t)p * OUT_PER;
    if (tid < DIMS) o[tid] = s_c1[tid];
    if (tid < 64) o[8 + tid] = s_c2[tid];
    for (int i = tid; i < 512; i += 128) o[72 + i] = s_c3[i];
    // c4 from WMMA accumulators: D layout lane<16 -> M=v, lane>=16 -> M=v+8
    {
        const int rbase = 16 * wv + 8 * (lane >> 4);
        float* o4 = o + 584;
#pragma unroll
        for (int v = 0; v < 8; ++v) {
            o4[(rbase + v) * 64 +  0 + ncol] = acc0[v];
            o4[(rbase + v) * 64 + 16 + ncol] = acc1[v];
            o4[(rbase + v) * 64 + 32 + ncol] = acc2[v];
            o4[(rbase + v) * 64 + 48 + ncol] = acc3[v];
        }
    }
}

extern "C" void kernel_launch(void* const* d_in, const int* in_sizes, int n_in,
                              void* d_out, int out_size, void* d_ws, size_t ws_size,
                              hipStream_t stream) {
    (void)n_in; (void)d_ws; (void)ws_size; (void)out_size;
    const float* path = (const float*)d_in[0];
    float* out = (float*)d_out;
    const int batch = in_sizes[0] / (PLEN * DIMS);   // 128
    sig_depth4_kernel<<<batch, 128, 0, stream>>>(path, out);
}